// ContrastDrivenFeatureAggregation_85117661872773
// MI455X (gfx1250) — compile-verified
//
#include <hip/hip_runtime.h>

typedef __attribute__((ext_vector_type(2))) float v2f;
typedef __attribute__((ext_vector_type(8))) float v8f;

namespace cdfa {
constexpr int BATCH = 4;
constexpr int CH    = 64;
constexpr int HH    = 128;
constexpr int WW    = 128;
constexpr int LL    = HH * WW;
constexpr int KK    = 9;
constexpr int AOUT  = 324;   // KK*KK*HEADS
constexpr int NA    = 36;    // HEADS*KK (i-reduced attention columns)
constexpr int NAP   = 48;    // padded to 3 WMMA n-tiles

// workspace layout (bytes)
constexpr size_t WS_WV  = 0;        // 4096 f32: W_v packed in WMMA-B fragment order
constexpr size_t WS_WFG = 16384;    // 3072 f32: reduced W_fg, B-fragment order
constexpr size_t WS_WBG = 28672;    // 3072 f32: reduced W_bg, B-fragment order
constexpr size_t WS_BR  = 40960;    // 48 f32: reduced bias
constexpr size_t WS_V   = 49152;    // B*L*C f32 value buffer, channel-last

constexpr int XS  = 136;  // LDS row stride for 64x128 input tile (16B-aligned, bank-decorrelated)
constexpr int AS  = 49;   // LDS row stride for 128x48 attention tile
constexpr int OTS = 132;  // LDS row stride for 64x128 output tile (16B-aligned rows)
}
using namespace cdfa;

// ---- gfx1250 async global<->LDS DMA (ASYNCcnt-tracked, no VGPR round trip) ----
__device__ __forceinline__ void async_ld_b128(unsigned lds_off, const float* gaddr) {
  asm volatile("global_load_async_to_lds_b128 %0, %1, off"
               :: "v"(lds_off), "v"(gaddr) : "memory");
}
__device__ __forceinline__ void async_st_b128(float* gaddr, unsigned lds_off) {
  asm volatile("global_store_async_from_lds_b128 %0, %1, off"
               :: "v"(gaddr), "v"(lds_off) : "memory");
}
__device__ __forceinline__ void wait_async0() {
  asm volatile("s_wait_asynccnt 0x0" ::: "memory");
}
__device__ __forceinline__ unsigned lds_off32(const void* p) {
  return (unsigned)(size_t)p;   // generic LDS pointer: low 32 bits are the LDS byte address
}

// ---------------------------------------------------------------------------
// Prep: pack W_v into f32 WMMA B-fragment order and fold the KKxKK -> KK
// column-sum (attn.sum(axis=i)) into the attention weights/bias, shrinking
// the attention GEMMs from N=324 to N=36 (padded 48).
// B-fragment layout (32-bit B, 4x16): lane<16 -> N=lane, K=0,1 in V0,V1;
// lane>=16 -> N=lane-16, K=2,3.  Packed index: ((nt*16+kt)*32+lane)*2+s.
// ---------------------------------------------------------------------------
__global__ void cdfa_prep(const float* __restrict__ Wv,
                          const float* __restrict__ Wfg, const float* __restrict__ bfg,
                          const float* __restrict__ Wbg, const float* __restrict__ bbg,
                          void* ws) {
  float* wvF  = (float*)((char*)ws + WS_WV);
  float* wfgF = (float*)((char*)ws + WS_WFG);
  float* wbgF = (float*)((char*)ws + WS_WBG);
  float* br   = (float*)((char*)ws + WS_BR);
  const int t = threadIdx.x;

  for (int idx = t; idx < 4 * 16 * 32 * 2; idx += 256) {   // W_v: nt(4) kt(16) lane(32) s(2)
    int s = idx & 1, lane = (idx >> 1) & 31, kt = (idx >> 6) & 15, nt = idx >> 10;
    int k = kt * 4 + (lane >> 4) * 2 + s;
    int n = nt * 16 + (lane & 15);
    wvF[idx] = Wv[k * CH + n];
  }
  for (int idx = t; idx < 3 * 16 * 32 * 2; idx += 256) {   // reduced attn weights
    int s = idx & 1, lane = (idx >> 1) & 31, kt = (idx >> 6) & 15, nt = idx >> 10;
    int k = kt * 4 + (lane >> 4) * 2 + s;
    int n = nt * 16 + (lane & 15);
    float sf = 0.f, sb = 0.f;
    if (n < NA) {
      int h = n / KK, j = n % KK;
      const float* wf = Wfg + k * AOUT + h * 81 + j;
      const float* wb = Wbg + k * AOUT + h * 81 + j;
#pragma unroll
      for (int i = 0; i < KK; ++i) { sf += wf[i * KK]; sb += wb[i * KK]; }
    }
    wfgF[idx] = sf;
    wbgF[idx] = sb;
  }
  if (t < NAP) {
    float s = 0.f;
    if (t < NA) {
      int h = t / KK, j = t % KK;
#pragma unroll
      for (int i = 0; i < KK; ++i) s += bfg[h * 81 + i * KK + j] + bbg[h * 81 + i * KK + j];
    }
    br[t] = s;
  }
}

// ---------------------------------------------------------------------------
// v = x_l @ W_v + b_v.  One block per image row (128 pixels); 8 waves, each
// owning a 16-row M-tile x full N=64.  K=64 -> 16 chained WMMA f32 16x16x4.
// A-fragment (32-bit A 16x4): lane<16: M=lane, V0/V1 = K0/K1; lane>=16: K2/K3.
// ---------------------------------------------------------------------------
__global__ void __launch_bounds__(256)
cdfa_value_gemm(const float* __restrict__ x, const float* __restrict__ bv, void* ws) {
  __shared__ __align__(16) float xt[CH * XS];
  const int b   = blockIdx.x >> 7;
  const int row = blockIdx.x & 127;
  const int m0  = row * WW;
  const int t   = threadIdx.x;

  // async-DMA the 64x128 x-tile into LDS (transpose-staged, 16B chunks)
  const float* xb = x + (size_t)b * CH * LL + m0;
  const unsigned xt0 = lds_off32(xt);
  for (int idx = t; idx < CH * WW / 4; idx += 256) {
    int k = idx >> 5, c4 = (idx & 31) * 4;
    async_ld_b128(xt0 + (unsigned)(k * XS + c4) * 4u, xb + (size_t)k * LL + c4);
  }
  wait_async0();
  __syncthreads();

  const int wave = t >> 5, lane = t & 31;
  const int half = lane >> 4, mrow = lane & 15;
  const float* wvF = (const float*)((const char*)ws + WS_WV);
  float* vout = (float*)((char*)ws + WS_V) + ((size_t)b * LL + m0 + wave * 16) * CH;

  v2f a[16];
#pragma unroll
  for (int kt = 0; kt < 16; ++kt) {
    int k = kt * 4 + half * 2;
    a[kt][0] = xt[(k + 0) * XS + wave * 16 + mrow];
    a[kt][1] = xt[(k + 1) * XS + wave * 16 + mrow];
  }

#pragma unroll
  for (int nt = 0; nt < 4; ++nt) {
    float bias = bv[nt * 16 + mrow];
    v8f c;
#pragma unroll
    for (int r = 0; r < 8; ++r) c[r] = bias;
#pragma unroll
    for (int kt = 0; kt < 16; ++kt) {
      v2f bf = ((const v2f*)wvF)[(nt * 16 + kt) * 32 + lane];
      c = __builtin_amdgcn_wmma_f32_16x16x4_f32(false, a[kt], false, bf,
                                                (short)0, c, false, false);
    }
#pragma unroll
    for (int r = 0; r < 8; ++r)                          // C: VGPR r -> M = r + half*8
      vout[(size_t)(r + half * 8) * CH + nt * 16 + mrow] = c[r];
  }
}

// ---------------------------------------------------------------------------
// attn = fg_l @ Wr_fg + bg_l @ Wr_bg + br (two GEMMs into one accumulator),
// then out[b, h*16+d, y, x] = sum_{ki,kj} attn[p, h*9+ki*3+kj] * v[y+ki-1, x+kj-1, h*16+d].
// ---------------------------------------------------------------------------
__global__ void __launch_bounds__(256)
cdfa_attn_agg(const float* __restrict__ fg, const float* __restrict__ bg,
              const void* __restrict__ ws, float* __restrict__ out) {
  __shared__ __align__(16) float smem[2 * CH * XS];   // fg/bg tiles; reused for attn + out tiles
  float* ft = smem;
  float* gt = smem + CH * XS;

  const int b   = blockIdx.x >> 7;
  const int row = blockIdx.x & 127;
  const int m0  = row * WW;
  const int t   = threadIdx.x;

  const float* fb = fg + (size_t)b * CH * LL + m0;
  const float* gb = bg + (size_t)b * CH * LL + m0;
  const unsigned ft0 = lds_off32(ft), gt0 = lds_off32(gt);
  for (int idx = t; idx < CH * WW / 4; idx += 256) {
    int k = idx >> 5, c4 = (idx & 31) * 4;
    unsigned loff = (unsigned)(k * XS + c4) * 4u;
    async_ld_b128(ft0 + loff, fb + (size_t)k * LL + c4);
    async_ld_b128(gt0 + loff, gb + (size_t)k * LL + c4);
  }
  wait_async0();
  __syncthreads();

  const int wave = t >> 5, lane = t & 31;
  const int half = lane >> 4, mrow = lane & 15;
  const float* wfgF = (const float*)((const char*)ws + WS_WFG);
  const float* wbgF = (const float*)((const char*)ws + WS_WBG);
  const float* br   = (const float*)((const char*)ws + WS_BR);

  v2f af[16], ag[16];
#pragma unroll
  for (int kt = 0; kt < 16; ++kt) {
    int k = kt * 4 + half * 2;
    af[kt][0] = ft[(k + 0) * XS + wave * 16 + mrow];
    af[kt][1] = ft[(k + 1) * XS + wave * 16 + mrow];
    ag[kt][0] = gt[(k + 0) * XS + wave * 16 + mrow];
    ag[kt][1] = gt[(k + 1) * XS + wave * 16 + mrow];
  }

  v8f c3[3];
#pragma unroll
  for (int nt = 0; nt < 3; ++nt) {
    float bias = br[nt * 16 + mrow];
    v8f c;
#pragma unroll
    for (int r = 0; r < 8; ++r) c[r] = bias;
#pragma unroll
    for (int kt = 0; kt < 16; ++kt) {
      v2f bwf = ((const v2f*)wfgF)[(nt * 16 + kt) * 32 + lane];
      c = __builtin_amdgcn_wmma_f32_16x16x4_f32(false, af[kt], false, bwf,
                                                (short)0, c, false, false);
      v2f bwg = ((const v2f*)wbgF)[(nt * 16 + kt) * 32 + lane];
      c = __builtin_amdgcn_wmma_f32_16x16x4_f32(false, ag[kt], false, bwg,
                                                (short)0, c, false, false);
    }
    c3[nt] = c;
  }
  __syncthreads();                       // done reading ft/gt -> reuse smem

  float* at = smem;                      // 128 x AS attention tile (6272 f32)
  float* ot = smem + WW * AS + 128;      // 64 x OTS output tile (ch-major, 16B rows)
#pragma unroll
  for (int nt = 0; nt < 3; ++nt)
#pragma unroll
    for (int r = 0; r < 8; ++r)
      at[(wave * 16 + half * 8 + r) * AS + nt * 16 + mrow] = c3[nt][r];
  __syncthreads();

  // 9-tap aggregation: thread -> (channel, pixel); v reads are 64-ch coalesced.
  const float* vb = (const float*)((const char*)ws + WS_V) + (size_t)b * LL * CH;
  const int ch = t & 63;
  const int hd = ch >> 4;
  for (int p = t >> 6; p < WW; p += 4) {
    float acc = 0.f;
#pragma unroll
    for (int ki = 0; ki < 3; ++ki) {
      int yy = row + ki - 1;
      if (yy < 0 || yy >= HH) continue;
#pragma unroll
      for (int kj = 0; kj < 3; ++kj) {
        int xx = p + kj - 1;
        if (xx < 0 || xx >= WW) continue;
        acc += at[p * AS + hd * KK + ki * 3 + kj] * vb[((size_t)yy * WW + xx) * CH + ch];
      }
    }
    ot[ch * OTS + p] = acc;
  }
  __syncthreads();

  // output tile is channel-major in LDS -> async-DMA straight to (B,C,H,W)
  float* ob = out + (size_t)b * CH * LL + m0;
  const unsigned ot0 = lds_off32(ot);
  for (int idx = t; idx < CH * WW / 4; idx += 256) {
    int c2 = idx >> 5, p4 = (idx & 31) * 4;
    async_st_b128(ob + (size_t)c2 * LL + p4, ot0 + (unsigned)(c2 * OTS + p4) * 4u);
  }
  // outstanding ASYNCcnt stores drain at S_ENDPGM's implicit wait-idle
}

extern "C" void kernel_launch(void* const* d_in, const int* in_sizes, int n_in,
                              void* d_out, int out_size, void* d_ws, size_t ws_size,
                              hipStream_t stream) {
  (void)in_sizes; (void)n_in; (void)out_size; (void)ws_size;
  const float* x   = (const float*)d_in[0];
  const float* fgp = (const float*)d_in[1];
  const float* bgp = (const float*)d_in[2];
  const float* Wv  = (const float*)d_in[3];
  const float* bv  = (const float*)d_in[4];
  const float* Wfg = (const float*)d_in[5];
  const float* bfg = (const float*)d_in[6];
  const float* Wbg = (const float*)d_in[7];
  const float* bbg = (const float*)d_in[8];

  cdfa_prep<<<1, 256, 0, stream>>>(Wv, Wfg, bfg, Wbg, bbg, d_ws);
  cdfa_value_gemm<<<BATCH * HH, 256, 0, stream>>>(x, bv, d_ws);
  cdfa_attn_agg<<<BATCH * HH, 256, 0, stream>>>(fgp, bgp, d_ws, (float*)d_out);
}